// interactiveGAT_16174846836806
// MI455X (gfx1250) — compile-verified
//
#include <hip/hip_runtime.h>
#include <math.h>

#define N_NODES 50000
#define E_EDGES 500000
#define D_DIM   256
#define H_HEADS 4
#define K_DIM   64          // D / H
#define L_LAYERS 2
#define NEG_SLOPE 0.2f

typedef __attribute__((ext_vector_type(2))) float v2f;
typedef __attribute__((ext_vector_type(8))) float v8f;
typedef __attribute__((ext_vector_type(4))) int   v4i;

#if defined(__has_builtin)
#  if __has_builtin(__builtin_amdgcn_global_load_async_to_lds_b128)
#    define HAVE_ASYNC_LDS 1
#  endif
#endif
#ifndef HAVE_ASYNC_LDS
#  define HAVE_ASYNC_LDS 0
#endif

typedef __attribute__((address_space(1))) v4i gv4i;   // global <4 x i32>
typedef __attribute__((address_space(3))) v4i lv4i;   // LDS    <4 x i32>

// ---------------- float ordered-key helpers for atomic max ----------------
__device__ __forceinline__ unsigned fkey(float f) {
    unsigned u = __float_as_uint(f);
    return (u & 0x80000000u) ? ~u : (u | 0x80000000u);
}
__device__ __forceinline__ float funkey(unsigned k) {
    return __uint_as_float((k & 0x80000000u) ? (k & 0x7FFFFFFFu) : ~k);
}

// ---------------- GEMM: xf[n, h*K+k] = sum_d x[n,d] * W[h,d,k] ------------
// Block (8 waves) computes a 16x256 slab: the 16x256 x-tile (16 KB,
// contiguous) is staged into LDS once (async-to-LDS when available), then
// each wave computes two 16x16 column tiles with V_WMMA_F32_16X16X4_F32,
// reusing each LDS A-fragment for two WMMAs.
// A layout (32-bit 16x4): lane l -> row = l&15, v0 = K(2*(l>>4)), v1 = +1.
// B layout (4x16) mirrors A: lane l -> col = l&15, v0 = K(2*(l>>4)), v1 = +1.
// C/D layout: vgpr i, lanes 0-15 -> M=i, lanes 16-31 -> M=i+8; N = l&15.
__global__ __launch_bounds__(256) void k_gemm(const float* __restrict__ x,
                                              const float* __restrict__ Wl, // (H,D,K)
                                              float* __restrict__ xf) {
    __shared__ float lds_x[16 * D_DIM];                // 16 KB
    const int t    = threadIdx.x;
    const int lane = t & 31;
    const int wave = t >> 5;                           // 0..7
    const int row0 = blockIdx.x * 16;                  // 3125 row tiles
    const int hi   = lane >> 4;                        // 0 or 1
    const int lo   = lane & 15;

    // ---- stage 16x256 x-tile (contiguous 16 KB) into LDS ----
    const float* __restrict__ gtile = x + (size_t)row0 * D_DIM;
#if HAVE_ASYNC_LDS
    {
        v4i* g0 = (v4i*)const_cast<float*>(gtile);
        v4i* l0 = (v4i*)lds_x;
        gv4i* g = (gv4i*)g0;
        lv4i* l = (lv4i*)l0;
        #pragma unroll
        for (int p = 0; p < 4; ++p) {                  // 256 thr * 16 B * 4 = 16 KB
            const int i = p * 256 + t;
            __builtin_amdgcn_global_load_async_to_lds_b128(g + i, l + i, 0, 0);
        }
        asm volatile("s_wait_asynccnt 0x0" ::: "memory");
    }
    __syncthreads();
#else
    {
        const float4* __restrict__ g = (const float4*)gtile;
        float4* s4 = (float4*)lds_x;
        #pragma unroll
        for (int p = 0; p < 4; ++p) s4[p * 256 + t] = g[p * 256 + t];
    }
    __syncthreads();
#endif

    // ---- two column tiles per wave: cols [32*wave, 32*wave+32) ----
    const int colA = wave * 32 + lo;
    const int colB = colA + 16;
    const float* __restrict__ wcolA = Wl + (size_t)(colA >> 6) * (D_DIM * K_DIM) + (colA & 63);
    const float* __restrict__ wcolB = Wl + (size_t)(colB >> 6) * (D_DIM * K_DIM) + (colB & 63);
    const float* __restrict__ arow  = lds_x + lo * D_DIM;

    v8f acc0 = {}, acc1 = {};
    #pragma unroll 4
    for (int kb = 0; kb < D_DIM; kb += 4) {
        const int ka = kb + 2 * hi;
        v2f a = *(const v2f*)(arow + ka);              // ds_load_b64
        v2f b0, b1;
        b0.x = wcolA[(size_t)ka * K_DIM];
        b0.y = wcolA[(size_t)(ka + 1) * K_DIM];
        b1.x = wcolB[(size_t)ka * K_DIM];
        b1.y = wcolB[(size_t)(ka + 1) * K_DIM];
        acc0 = __builtin_amdgcn_wmma_f32_16x16x4_f32(false, a, false, b0,
                                                     (short)0, acc0, false, false);
        acc1 = __builtin_amdgcn_wmma_f32_16x16x4_f32(false, a, false, b1,
                                                     (short)0, acc1, false, false);
    }

    float* __restrict__ out0 = xf + (size_t)(row0 + 8 * hi) * D_DIM + colA;
    #pragma unroll
    for (int i = 0; i < 8; ++i) out0[(size_t)i * D_DIM] = acc0[i];
    float* __restrict__ out1 = xf + (size_t)(row0 + 8 * hi) * D_DIM + colB;
    #pragma unroll
    for (int i = 0; i < 8; ++i) out1[(size_t)i * D_DIM] = acc1[i];
}

// ---------------- attention logits per (node, head) -----------------------
__global__ void k_alpha(const float* __restrict__ xf, const float* __restrict__ Al,
                        float* __restrict__ asrc, float* __restrict__ adst) {
    int idx = blockIdx.x * blockDim.x + threadIdx.x;   // n*H + h
    if (idx >= N_NODES * H_HEADS) return;
    int n = idx >> 2, h = idx & 3;
    const float* __restrict__ v  = xf + (size_t)n * D_DIM + h * K_DIM;
    const float* __restrict__ as = Al + h * (2 * K_DIM);
    const float* __restrict__ ad = as + K_DIM;
    float s0 = 0.f, s1 = 0.f;
    #pragma unroll 8
    for (int k = 0; k < K_DIM; ++k) { float t = v[k]; s0 += t * as[k]; s1 += t * ad[k]; }
    asrc[idx] = s0;
    adst[idx] = s1;
}

// ---------------- edge pass 1: leaky relu score + segment max -------------
__global__ void k_edge_max(const int* __restrict__ src, const int* __restrict__ dst,
                           const float* __restrict__ asrc, const float* __restrict__ adst,
                           float* __restrict__ sbuf, unsigned* __restrict__ mkey) {
    int idx = blockIdx.x * blockDim.x + threadIdx.x;   // e*H + h
    if (idx >= E_EDGES * H_HEADS) return;
    int e = idx >> 2, h = idx & 3;
    int sN = src[e], dN = dst[e];
    float s = asrc[sN * H_HEADS + h] + adst[dN * H_HEADS + h];
    s = (s >= 0.f) ? s : NEG_SLOPE * s;
    sbuf[idx] = s;
    atomicMax(mkey + dN * H_HEADS + h, fkey(s));
}

// ---------------- edge pass 2: exp(s - m) + segment sum -------------------
__global__ void k_edge_exp(const int* __restrict__ dst,
                           float* __restrict__ sbuf, const unsigned* __restrict__ mkey,
                           float* __restrict__ denom) {
    int idx = blockIdx.x * blockDim.x + threadIdx.x;   // e*H + h
    if (idx >= E_EDGES * H_HEADS) return;
    int e = idx >> 2, h = idx & 3;
    int dN = dst[e];
    float m  = funkey(mkey[dN * H_HEADS + h]);
    float ev = __expf(sbuf[idx] - m);
    sbuf[idx] = ev;
    atomicAdd(denom + dN * H_HEADS + h, ev);
}

// ---------------- edge pass 3: scatter normalized messages ----------------
// One wave per edge: 256 columns in 8 coalesced strides of 32.
__global__ __launch_bounds__(256) void k_edge_msg(const int* __restrict__ src,
                                                  const int* __restrict__ dst,
                                                  const float* __restrict__ xf,
                                                  const float* __restrict__ sbuf,
                                                  const float* __restrict__ denom,
                                                  float* __restrict__ outAcc) {
    int e = blockIdx.x * 8 + (threadIdx.x >> 5);
    if (e >= E_EDGES) return;
    int lane = threadIdx.x & 31;
    int sN = src[e], dN = dst[e];
    const float* __restrict__ xrow = xf + (size_t)sN * D_DIM;
    float* __restrict__ orow = outAcc + (size_t)dN * D_DIM;
    float norms[H_HEADS];
    #pragma unroll
    for (int h = 0; h < H_HEADS; ++h)
        norms[h] = sbuf[(size_t)e * H_HEADS + h] / denom[dN * H_HEADS + h];
    #pragma unroll
    for (int i = 0; i < 8; ++i) {
        int c = lane + i * 32;          // columns [32i, 32i+31] -> head = i>>1 (static)
        atomicAdd(orow + c, norms[i >> 1] * xrow[c]);
    }
}

// ---------------- combine: x_next = x + elu(out + bias) -------------------
__global__ void k_combine(const float* __restrict__ xcur, const float* __restrict__ outAcc,
                          const float* __restrict__ bias, float* __restrict__ xnext) {
    int idx = blockIdx.x * blockDim.x + threadIdx.x;   // n*D + c
    if (idx >= N_NODES * D_DIM) return;
    float o = outAcc[idx] + bias[idx & (D_DIM - 1)];
    float el = (o > 0.f) ? o : (__expf(o) - 1.0f);
    xnext[idx] = xcur[idx] + el;
}

// --------------------------------------------------------------------------
extern "C" void kernel_launch(void* const* d_in, const int* in_sizes, int n_in,
                              void* d_out, int out_size, void* d_ws, size_t ws_size,
                              hipStream_t stream) {
    const float* x   = (const float*)d_in[0];          // (N, D)
    const int*   ei  = (const int*)d_in[1];            // (2, E)
    const float* W   = (const float*)d_in[5];          // (L, H, D, K)
    const float* A   = (const float*)d_in[6];          // (L, H, 2K)
    const float* Bb  = (const float*)d_in[7];          // (L, D)
    float* out = (float*)d_out;                        // (N, D)

    const int* src = ei;
    const int* dst = ei + E_EDGES;

    // workspace partition (bytes)
    char* ws = (char*)d_ws;
    const size_t szND = (size_t)N_NODES * D_DIM * sizeof(float);   // 51.2 MB
    const size_t szNH = (size_t)N_NODES * H_HEADS * sizeof(float); // 0.8 MB
    const size_t szEH = (size_t)E_EDGES * H_HEADS * sizeof(float); // 8 MB
    float*    xf     = (float*)(ws);                 ws += szND;
    float*    outAcc = (float*)(ws);                 ws += szND;
    float*    xbuf   = (float*)(ws);                 ws += szND;
    float*    asrc   = (float*)(ws);                 ws += szNH;
    float*    adst   = (float*)(ws);                 ws += szNH;
    unsigned* mkey   = (unsigned*)(ws);              ws += szNH;
    float*    denom  = (float*)(ws);                 ws += szNH;
    float*    sbuf   = (float*)(ws);                 ws += szEH;

    const int gemmBlocks = N_NODES / 16;             // 3125, 8 waves/block
    const int nhBlocks   = (N_NODES * H_HEADS + 255) / 256;
    const int ehBlocks   = (E_EDGES * H_HEADS + 255) / 256;
    const int msgBlocks  = (E_EDGES + 7) / 8;
    const int ndBlocks   = (N_NODES * D_DIM + 255) / 256;

    const float* xcur = x;
    for (int l = 0; l < L_LAYERS; ++l) {
        (void)hipMemsetAsync(mkey,   0, szNH, stream);   // key 0 == -inf ordering
        (void)hipMemsetAsync(denom,  0, szNH, stream);
        (void)hipMemsetAsync(outAcc, 0, szND, stream);

        const float* Wl = W  + (size_t)l * H_HEADS * D_DIM * K_DIM;
        const float* Al = A  + (size_t)l * H_HEADS * 2 * K_DIM;
        const float* bl = Bb + (size_t)l * D_DIM;

        k_gemm<<<gemmBlocks, 256, 0, stream>>>(xcur, Wl, xf);
        k_alpha<<<nhBlocks, 256, 0, stream>>>(xf, Al, asrc, adst);
        k_edge_max<<<ehBlocks, 256, 0, stream>>>(src, dst, asrc, adst, sbuf, mkey);
        k_edge_exp<<<ehBlocks, 256, 0, stream>>>(dst, sbuf, mkey, denom);
        k_edge_msg<<<msgBlocks, 256, 0, stream>>>(src, dst, xf, sbuf, denom, outAcc);

        float* xnext = (l == L_LAYERS - 1) ? out : xbuf;
        k_combine<<<ndBlocks, 256, 0, stream>>>(xcur, outAcc, bl, xnext);
        xcur = xnext;
    }
}